// ST_Masked_Encoder_layer_73083163509002
// MI455X (gfx1250) — compile-verified
//
#include <hip/hip_runtime.h>
#include <hip/hip_bf16.h>

typedef float v2f __attribute__((ext_vector_type(2)));
typedef float v8f __attribute__((ext_vector_type(8)));

constexpr int kB  = 32, kT = 12, kN = 62, kF = 38, kH = 24;
constexpr int kDM = 256, kNH = 4, kDK = 64;
constexpr int kNF  = kN * kF;              // 2356
constexpr int kCIN = kH * kNF;             // 56544
constexpr long kSBT = (long)kB * kT * kNF; // 904704
constexpr long kSH  = (long)kB * kH * kNF; // 1809408
constexpr int kJ = kB * kT;                // 384 GEMM rows
constexpr int kNIter = (kCIN / 4) / 8;     // 1767 split-K iters per wave

__device__ __forceinline__ float sigm(float x) { return 1.f / (1.f + __expf(-x)); }

// ---- CDNA5 f32 WMMA: D(16x16) += A(16x4) * B(4x16), exact fp32 ----
__device__ __forceinline__ v8f wmma4(v2f a, v2f b, v8f c) {
  return __builtin_amdgcn_wmma_f32_16x16x4_f32(false, a, false, b, (short)0, c, false, false);
}

// ------------------------------ utility ------------------------------
__global__ void k_zero(float* p, long n) {
  long i = (long)blockIdx.x * blockDim.x + threadIdx.x;
  long st = (long)gridDim.x * blockDim.x;
  for (; i < n; i += st) p[i] = 0.f;
}
__global__ void k_copy(float* d, const float* s, long n) {
  long i = (long)blockIdx.x * blockDim.x + threadIdx.x;
  long st = (long)gridDim.x * blockDim.x;
  for (; i < n; i += st) d[i] = s[i];
}

// ------------------------------ CBAM ------------------------------
__global__ void k_conv3x3(const float* __restrict__ src, long srcB, int Cin,
                          const float* __restrict__ w, const float* __restrict__ bias,
                          float* __restrict__ dst, int Cout, int doRelu) {
  int idx = blockIdx.x * blockDim.x + threadIdx.x;
  int total = kB * Cout * kNF;
  if (idx >= total) return;
  int p = idx % kNF;
  int co = (idx / kNF) % Cout;
  int b = idx / (kNF * Cout);
  int n = p / kF, f = p % kF;
  float acc = bias[co];
  for (int ci = 0; ci < Cin; ++ci) {
    const float* sp = src + (long)b * srcB + (long)ci * kNF;
    const float* wp = w + ((long)co * Cin + ci) * 9;
    for (int ky = 0; ky < 3; ++ky) {
      int nn = n + ky - 1;
      if ((unsigned)nn >= (unsigned)kN) continue;
      for (int kx = 0; kx < 3; ++kx) {
        int ff = f + kx - 1;
        if ((unsigned)ff >= (unsigned)kF) continue;
        acc += wp[ky * 3 + kx] * sp[nn * kF + ff];
      }
    }
  }
  if (doRelu) acc = fmaxf(acc, 0.f);
  dst[((long)b * Cout + co) * kNF + p] = acc;
}

__global__ void k_chanstat(const float* __restrict__ x, float* __restrict__ mean,
                           float* __restrict__ mx) {
  int bc = blockIdx.x;  // b*12+c
  const float* p = x + (long)bc * kNF;
  float s = 0.f, m = -3.4e38f;
  for (int i = threadIdx.x; i < kNF; i += 256) { float v = p[i]; s += v; m = fmaxf(m, v); }
  __shared__ float ss[256], sm[256];
  ss[threadIdx.x] = s; sm[threadIdx.x] = m;
  __syncthreads();
  for (int st = 128; st > 0; st >>= 1) {
    if (threadIdx.x < st) {
      ss[threadIdx.x] += ss[threadIdx.x + st];
      sm[threadIdx.x] = fmaxf(sm[threadIdx.x], sm[threadIdx.x + st]);
    }
    __syncthreads();
  }
  if (threadIdx.x == 0) { mean[bc] = ss[0] / (float)kNF; mx[bc] = sm[0]; }
}

__global__ void k_chanattn(const float* __restrict__ mean, const float* __restrict__ mx,
                           const float* __restrict__ w1, const float* __restrict__ b1,
                           const float* __restrict__ w2, const float* __restrict__ b2,
                           float* __restrict__ ca) {
  int idx = blockIdx.x * blockDim.x + threadIdx.x;
  if (idx >= kB * kT) return;
  int b = idx / kT, c = idx % kT;
  const float* v1 = mean + b * kT;
  const float* v2 = mx + b * kT;
  float acc = 2.f * b2[c];
  for (int j = 0; j < 3; ++j) {
    float h1 = b1[j], h2 = b1[j];
    for (int i = 0; i < kT; ++i) { h1 += w1[j * kT + i] * v1[i]; h2 += w1[j * kT + i] * v2[i]; }
    acc += w2[c * 3 + j] * (fmaxf(h1, 0.f) + fmaxf(h2, 0.f));
  }
  ca[idx] = sigm(acc);
}

__global__ void k_scalestats(float* __restrict__ x, const float* __restrict__ ca,
                             float* __restrict__ s) {
  int idx = blockIdx.x * blockDim.x + threadIdx.x;
  if (idx >= kB * kNF) return;
  int b = idx / kNF, p = idx % kNF;
  float mean = 0.f, mx = -3.4e38f;
  for (int c = 0; c < kT; ++c) {
    long o = ((long)b * kT + c) * kNF + p;
    float v = x[o] * ca[b * kT + c];
    x[o] = v;
    mean += v; mx = fmaxf(mx, v);
  }
  s[(long)b * 2 * kNF + p] = mean / (float)kT;
  s[(long)b * 2 * kNF + kNF + p] = mx;
}

__global__ void k_spatial(float* __restrict__ x, const float* __restrict__ s,
                          const float* __restrict__ w, const float* __restrict__ bias) {
  int idx = blockIdx.x * blockDim.x + threadIdx.x;
  if (idx >= kB * kNF) return;
  int b = idx / kNF, p = idx % kNF;
  int n = p / kF, f = p % kF;
  float acc = bias[0];
  for (int c = 0; c < 2; ++c) {
    const float* sp = s + (long)b * 2 * kNF + (long)c * kNF;
    for (int ky = 0; ky < 7; ++ky) {
      int nn = n + ky - 3;
      if ((unsigned)nn >= (unsigned)kN) continue;
      for (int kx = 0; kx < 7; ++kx) {
        int ff = f + kx - 3;
        if ((unsigned)ff >= (unsigned)kF) continue;
        acc += w[(c * 7 + ky) * 7 + kx] * sp[nn * kF + ff];
      }
    }
  }
  float sa = sigm(acc);
  for (int c = 0; c < kT; ++c) x[((long)b * kT + c) * kNF + p] *= sa;
}

// ------------------------------ GCLSTM ------------------------------
// zg[b,c,n,f] = sum_j adj[n,j] * zc[b,c,j,f], zc = concat(x[Cx ch], h[24 ch])
__global__ void k_gcadj(const float* __restrict__ xp, long xB, int Cx,
                        const float* __restrict__ hp, const float* __restrict__ adj,
                        float* __restrict__ zg, int Ctot) {
  int idx = blockIdx.x * blockDim.x + threadIdx.x;
  int total = kB * Ctot * kNF;
  if (idx >= total) return;
  int p = idx % kNF;
  int c = (idx / kNF) % Ctot;
  int b = idx / (kNF * Ctot);
  int n = p / kF, f = p % kF;
  const float* src = (c < Cx) ? (xp + (long)b * xB + (long)c * kNF)
                              : (hp + ((long)b * kH + (c - Cx)) * kNF);
  const float* ar = adj + n * kN;
  float acc = 0.f;
  for (int j = 0; j < kN; ++j) acc += ar[j] * src[j * kF + f];
  zg[idx] = acc;
}

// g = W[96,C] * zg + b ; LSTM gates; optional history write (layer-1)
__global__ void k_gcgates(const float* __restrict__ zg, int C,
                          const float* __restrict__ W, const float* __restrict__ bias,
                          float* __restrict__ h, float* __restrict__ c,
                          float* __restrict__ hist, int t) {
  __shared__ float sW[96 * 48];
  __shared__ float sB[96];
  for (int i = threadIdx.x; i < 96 * C; i += blockDim.x) sW[i] = W[i];
  for (int i = threadIdx.x; i < 96; i += blockDim.x) sB[i] = bias[i];
  __syncthreads();
  long total = kSH;
  for (long idx = (long)blockIdx.x * blockDim.x + threadIdx.x; idx < total;
       idx += (long)gridDim.x * blockDim.x) {
    int p = (int)(idx % kNF);
    int hh = (int)((idx / kNF) % kH);
    int b = (int)(idx / ((long)kNF * kH));
    float a0 = 0.f, a1 = 0.f, a2 = 0.f, a3 = 0.f;
    const float* zp = zg + (long)b * C * kNF + p;
    for (int cc = 0; cc < C; ++cc) {
      float zv = zp[(long)cc * kNF];
      a0 += sW[(0 * kH + hh) * C + cc] * zv;
      a1 += sW[(1 * kH + hh) * C + cc] * zv;
      a2 += sW[(2 * kH + hh) * C + cc] * zv;
      a3 += sW[(3 * kH + hh) * C + cc] * zv;
    }
    float ig = sigm(a0 + sB[hh]);
    float fg = sigm(a1 + sB[kH + hh]);
    float gg = tanhf(a2 + sB[2 * kH + hh]);
    float og = sigm(a3 + sB[3 * kH + hh]);
    float cn = fg * c[idx] + ig * gg;
    float hn = og * tanhf(cn);
    c[idx] = cn;
    h[idx] = hn;
    if (hist) hist[idx * kT + t] = hn;  // idx == b*CIN + hh*NF + p
  }
}

// h_t/c_t = conv2d(concat(ha,hb), fc_w[24,48,3,3]) + fc_b
__global__ void k_fcconv(const float* __restrict__ ha, const float* __restrict__ hb,
                         const float* __restrict__ w, const float* __restrict__ bias,
                         float* __restrict__ dst) {
  int idx = blockIdx.x * blockDim.x + threadIdx.x;
  int total = kB * kH * kNF;
  if (idx >= total) return;
  int p = idx % kNF;
  int co = (idx / kNF) % kH;
  int b = idx / (kNF * kH);
  int n = p / kF, f = p % kF;
  float acc = bias[co];
  for (int ci = 0; ci < 48; ++ci) {
    const float* sp = (ci < kH) ? (ha + ((long)b * kH + ci) * kNF)
                                : (hb + ((long)b * kH + ci - kH) * kNF);
    const float* wp = w + ((long)co * 48 + ci) * 9;
    for (int ky = 0; ky < 3; ++ky) {
      int nn = n + ky - 1;
      if ((unsigned)nn >= (unsigned)kN) continue;
      for (int kx = 0; kx < 3; ++kx) {
        int ff = f + kx - 1;
        if ((unsigned)ff >= (unsigned)kF) continue;
        acc += wp[ky * 3 + kx] * sp[nn * kF + ff];
      }
    }
  }
  dst[((long)b * kH + co) * kNF + p] = acc;
}

// --------------------- emb conv1d as WMMA GEMM ---------------------
// out[b*T+t, dm] = sum_kk sum_cin emb_w[dm,cin,kk] * hist[b,cin,t+kk-1] + emb_b[dm]
// Block = 8 waves, one 16x16 D tile, split-K across waves + LDS reduce.
// Split-K trip count is a compile-time constant (kNIter) so the hot loop is a
// uniform scalar-branch loop: EXEC stays all-1s around every v_wmma by construction.
__global__ void __launch_bounds__(256) k_embgemm(const float* __restrict__ W,
                                                 const float* __restrict__ bias,
                                                 const float* __restrict__ hist,
                                                 float* __restrict__ out) {
  __shared__ float red[8][32][8];
  int lane = threadIdx.x & 31, wave = threadIdx.x >> 5;
  int tileM = blockIdx.x & 15;   // 16 tiles over DM=256
  int tileN = blockIdx.x >> 4;   // 24 tiles over B*T=384
  int l15 = lane & 15, kHi = lane >> 4;
  int iA = tileM * 16 + l15;           // dm row of A
  int col = tileN * 16 + l15;          // output row (b*T+t)
  int bb = col / kT, tl = col % kT;
  v8f acc = {0.f, 0.f, 0.f, 0.f, 0.f, 0.f, 0.f, 0.f};
  for (int kk = 0; kk < 3; ++kk) {
    int tt = tl + kk - 1;
    bool valid = (unsigned)tt < (unsigned)kT;
    const float* hb = hist + (long)bb * kCIN * kT + (valid ? tt : 0);
    // this wave starts at cin = wave*4; advances by 32 cin per iteration
    long aOff = ((long)iA * kCIN + wave * 4 + kHi * 2) * 3 + kk;
    long bOff = (long)(wave * 4 + kHi) * kT;
#pragma unroll 3
    for (int it = 0; it < kNIter; ++it) {
      v2f a, bf;
      a.x = W[aOff];
      a.y = W[aOff + 3];
      float bx = hb[bOff];
      float by = hb[bOff + 2 * kT];
      bf.x = valid ? bx : 0.f;
      bf.y = valid ? by : 0.f;
      acc = wmma4(a, bf, acc);
      aOff += 96;        // 8 waves * 4 cin * 3 floats
      bOff += 32 * kT;   // 8 waves * 4 cin rows
    }
  }
#pragma unroll
  for (int r = 0; r < 8; ++r) red[wave][lane][r] = acc[r];
  __syncthreads();
  if (wave == 0) {
#pragma unroll
    for (int r = 0; r < 8; ++r) {
      float s2 = 0.f;
      for (int w2 = 0; w2 < 8; ++w2) s2 += red[w2][lane][r];
      int i = tileM * 16 + r + (kHi << 3);   // D: M = r (+8 for upper lanes)
      int j = tileN * 16 + l15;              // D: N = lane&15
      out[(long)j * kDM + i] = s2 + bias[i];
    }
  }
}

// --------------- generic projection GEMM: D[j,i] = X[j,:]·Wm[i,:] ---------------
// K is compile-time (256 or 128); tile selection is scalarized via readfirstlane
// so the guard branch around the WMMA loop is SALU and EXEC stays all-1s.
template <int K>
__global__ void __launch_bounds__(256) k_proj(const float* __restrict__ X,
                                              const float* __restrict__ Wm, int M,
                                              const float* __restrict__ bias,
                                              const float* __restrict__ res, int resLd,
                                              float* __restrict__ out, int outLd,
                                              int doRelu, int perBDst, int segOff) {
  int lane = threadIdx.x & 31;
  int wave = __builtin_amdgcn_readfirstlane(threadIdx.x >> 5);
  int MT = (M + 15) >> 4;
  int JT = kJ >> 4;  // 24
  int tile = blockIdx.x * 8 + wave;
  if (tile >= MT * JT) return;
  int tileM = tile % MT, tileJ = tile / MT;
  int l15 = lane & 15, kHi = lane >> 4;
  int iA = tileM * 16 + l15;
  int jB = tileJ * 16 + l15;
  bool iOk = iA < M;
  const float* wrow = Wm + (long)(iOk ? iA : 0) * K;
  const float* xrow = X + (long)jB * K;
  v8f acc = {0.f, 0.f, 0.f, 0.f, 0.f, 0.f, 0.f, 0.f};
#pragma unroll 8
  for (int k0 = 0; k0 < K; k0 += 4) {
    v2f a, bf;
    a.x = iOk ? wrow[k0 + kHi * 2] : 0.f;
    a.y = iOk ? wrow[k0 + kHi * 2 + 1] : 0.f;
    bf.x = xrow[k0 + kHi];
    bf.y = xrow[k0 + 2 + kHi];
    acc = wmma4(a, bf, acc);
  }
#pragma unroll
  for (int r = 0; r < 8; ++r) {
    int i = tileM * 16 + r + (kHi << 3);
    if (i >= M) continue;
    int j = tileJ * 16 + l15;
    int dj = (j / kT) * perBDst + segOff + (j % kT);
    float v = acc[r] + (bias ? bias[i] : 0.f);
    if (doRelu) v = fmaxf(v, 0.f);
    if (res) v += res[(long)dj * resLd + i];
    out[(long)dj * outLd + i] = v;
  }
}

// ------------------------------ attention ------------------------------
__global__ void k_attn(const float* __restrict__ q, const float* __restrict__ k,
                       const float* __restrict__ v, float* __restrict__ ctx) {
  __shared__ float sS[kT * 3 * kT];  // 12x36
  int b = blockIdx.x / kNH, h = blockIdx.x % kNH;
  int tid = threadIdx.x;
  for (int idx = tid; idx < kT * 36; idx += blockDim.x) {
    int qt = idx / 36, kt = idx % 36;
    const float* qp = q + (long)(b * kT + qt) * kDM + h * kDK;
    const float* kp = k + (long)(b * 36 + kt) * kDM + h * kDK;
    float d = 0.f;
    for (int e = 0; e < kDK; ++e) d += qp[e] * kp[e];
    sS[idx] = d * 0.125f;  // 1/sqrt(64)
  }
  __syncthreads();
  if (tid < kT) {
    float* row = sS + tid * 36;
    float m = -3.4e38f;
    for (int i = 0; i < 36; ++i) m = fmaxf(m, row[i]);
    float s = 0.f;
    for (int i = 0; i < 36; ++i) { row[i] = __expf(row[i] - m); s += row[i]; }
    float inv = 1.f / s;
    for (int i = 0; i < 36; ++i) row[i] *= inv;
  }
  __syncthreads();
  for (int idx = tid; idx < kT * kDK; idx += blockDim.x) {
    int qt = idx / kDK, e = idx % kDK;
    float acc = 0.f;
    for (int kt = 0; kt < 36; ++kt)
      acc += sS[qt * 36 + kt] * v[(long)(b * 36 + kt) * kDM + h * kDK + e];
    ctx[(long)(b * kT + qt) * kDM + h * kDK + e] = acc;
  }
}

__global__ void k_ln(const float* __restrict__ x, float* __restrict__ y) {
  __shared__ float sh[kDM];
  int j = blockIdx.x, tid = threadIdx.x;
  float v = x[(long)j * kDM + tid];
  sh[tid] = v;
  __syncthreads();
  for (int st = 128; st > 0; st >>= 1) {
    if (tid < st) sh[tid] += sh[tid + st];
    __syncthreads();
  }
  float mean = sh[0] / (float)kDM;
  __syncthreads();
  float d = v - mean;
  sh[tid] = d * d;
  __syncthreads();
  for (int st = 128; st > 0; st >>= 1) {
    if (tid < st) sh[tid] += sh[tid + st];
    __syncthreads();
  }
  float var = sh[0] / (float)kDM;
  y[(long)j * kDM + tid] = d * rsqrtf(var + 1e-5f);
}

// ------------------------------ driver ------------------------------
extern "C" void kernel_launch(void* const* d_in, const int* in_sizes, int n_in,
                              void* d_out, int out_size, void* d_ws, size_t ws_size,
                              hipStream_t stream) {
  const float* ODFlow = (const float*)d_in[0];
  const float* COD    = (const float*)d_in[1];
  const float* adj    = (const float*)d_in[2];
  const float* hc_w = (const float*)d_in[3];  const float* hc_b = (const float*)d_in[4];
  const float* hca_w1 = (const float*)d_in[5]; const float* hca_b1 = (const float*)d_in[6];
  const float* hca_w2 = (const float*)d_in[7]; const float* hca_b2 = (const float*)d_in[8];
  const float* hsa_w = (const float*)d_in[9];  const float* hsa_b = (const float*)d_in[10];
  const float* tc_w = (const float*)d_in[11]; const float* tc_b = (const float*)d_in[12];
  const float* tca_w1 = (const float*)d_in[13]; const float* tca_b1 = (const float*)d_in[14];
  const float* tca_w2 = (const float*)d_in[15]; const float* tca_b2 = (const float*)d_in[16];
  const float* tsa_w = (const float*)d_in[17]; const float* tsa_b = (const float*)d_in[18];
  const float* lw0 = (const float*)d_in[19]; const float* lb0 = (const float*)d_in[20];
  const float* lw1 = (const float*)d_in[21]; const float* lb1 = (const float*)d_in[22];
  const float* fc_w = (const float*)d_in[23]; const float* fc_b = (const float*)d_in[24];
  const float* emb_w = (const float*)d_in[25]; const float* emb_b = (const float*)d_in[26];
  const float* aq_w = (const float*)d_in[27]; const float* aq_b = (const float*)d_in[28];
  const float* ak_w = (const float*)d_in[29]; const float* ak_b = (const float*)d_in[30];
  const float* av_w = (const float*)d_in[31]; const float* av_b = (const float*)d_in[32];
  const float* ao_w = (const float*)d_in[33]; const float* ao_b = (const float*)d_in[34];
  const float* f1_w = (const float*)d_in[35]; const float* f2_w = (const float*)d_in[36];
  const float* l1_w = (const float*)d_in[37]; const float* l1_b = (const float*)d_in[38];
  float* ws = (float*)d_ws;
  float* outp = (float*)d_out;
  (void)in_sizes; (void)n_in; (void)out_size; (void)ws_size;

  // workspace layout (floats)
  size_t off = 0;
  auto A = [&](size_t n) { size_t r = off; off += n; return r; };
  const size_t o_cbw = A(kSBT), o_cbd = A(kSBT), o_cbt = A(kSBT);
  const size_t o_cam = A(kB * kT), o_cax = A(kB * kT), o_ca = A(kB * kT);
  const size_t o_s = A((size_t)kB * 2 * kNF);
  const size_t o_h0 = A(kSH), o_c0 = A(kSH), o_h1 = A(kSH), o_c1 = A(kSH);
  const size_t o_hw = A(kSH), o_cw = A(kSH), o_hd = A(kSH), o_cd = A(kSH);
  const size_t o_ht = A(kSH), o_ct = A(kSH);
  const size_t o_zg = A((size_t)kB * 48 * kNF);
  const size_t o_hsw = A((size_t)kSH * kT), o_hsd = A((size_t)kSH * kT), o_hst = A((size_t)kSH * kT);
  const size_t o_qin = A(384 * 256), o_din = A(384 * 256), o_win = A(384 * 256);
  const size_t o_q = A(384 * 256), o_k = A(1152 * 256), o_v = A(1152 * 256);
  const size_t o_ctx = A(384 * 256), o_ctx2 = A(384 * 256);
  const size_t o_t1 = A(384 * 128), o_ctx3 = A(384 * 256);

  // ---- CBAM x3 ----
  auto run_cbam = [&](const float* src, long srcB, int Cin, const float* cw, const float* cb,
                      const float* w1, const float* b1, const float* w2, const float* b2,
                      const float* sw, const float* sb, float* dst) {
    int total = kB * kT * kNF;
    k_conv3x3<<<(total + 255) / 256, 256, 0, stream>>>(src, srcB, Cin, cw, cb, dst, kT, 1);
    k_chanstat<<<kB * kT, 256, 0, stream>>>(dst, ws + o_cam, ws + o_cax);
    k_chanattn<<<(kB * kT + 127) / 128, 128, 0, stream>>>(ws + o_cam, ws + o_cax, w1, b1, w2, b2,
                                                          ws + o_ca);
    k_scalestats<<<(kB * kNF + 255) / 256, 256, 0, stream>>>(dst, ws + o_ca, ws + o_s);
    k_spatial<<<(kB * kNF + 255) / 256, 256, 0, stream>>>(dst, ws + o_s, sw, sb);
  };
  run_cbam(ODFlow,       2 * kNF, 1, hc_w, hc_b, hca_w1, hca_b1, hca_w2, hca_b2, hsa_w, hsa_b, ws + o_cbw);
  run_cbam(ODFlow + kNF, 2 * kNF, 1, hc_w, hc_b, hca_w1, hca_b1, hca_w2, hca_b2, hsa_w, hsa_b, ws + o_cbd);
  run_cbam(COD, (long)kT * kNF, kT, tc_w, tc_b, tca_w1, tca_b1, tca_w2, tca_b2, tsa_w, tsa_b, ws + o_cbt);

  // ---- GCLSTM runs ----
  float *h0 = ws + o_h0, *c0 = ws + o_c0, *h1 = ws + o_h1, *c1 = ws + o_c1, *zg = ws + o_zg;
  int gatesGrid = (int)((kSH + 255) / 256);
  auto run_gclstm = [&](const float* xseq, float* hist, const float* ih, const float* ic) {
    if (ih) {
      k_copy<<<1024, 256, 0, stream>>>(h0, ih, kSH);
      k_copy<<<1024, 256, 0, stream>>>(c0, ic, kSH);
      k_copy<<<1024, 256, 0, stream>>>(h1, ih, kSH);
      k_copy<<<1024, 256, 0, stream>>>(c1, ic, kSH);
    } else {
      k_zero<<<2048, 256, 0, stream>>>(h0, 4 * kSH);  // h0,c0,h1,c1 contiguous
    }
    for (int t = 0; t < kT; ++t) {
      int t0 = kB * 25 * kNF, t1n = kB * 48 * kNF;
      k_gcadj<<<(t0 + 255) / 256, 256, 0, stream>>>(xseq + (long)t * kNF, (long)kT * kNF, 1, h0,
                                                    adj, zg, 25);
      k_gcgates<<<gatesGrid, 256, 0, stream>>>(zg, 25, lw0, lb0, h0, c0, nullptr, t);
      k_gcadj<<<(t1n + 255) / 256, 256, 0, stream>>>(h0, (long)kH * kNF, kH, h1, adj, zg, 48);
      k_gcgates<<<gatesGrid, 256, 0, stream>>>(zg, 48, lw1, lb1, h1, c1, hist, t);
    }
  };
  run_gclstm(ws + o_cbw, ws + o_hsw, nullptr, nullptr);
  k_copy<<<1024, 256, 0, stream>>>(ws + o_hw, h1, kSH);
  k_copy<<<1024, 256, 0, stream>>>(ws + o_cw, c1, kSH);
  run_gclstm(ws + o_cbd, ws + o_hsd, nullptr, nullptr);
  k_copy<<<1024, 256, 0, stream>>>(ws + o_hd, h1, kSH);
  k_copy<<<1024, 256, 0, stream>>>(ws + o_cd, c1, kSH);
  // h_t / c_t via fc conv on concatenated states
  {
    int total = (int)kSH;
    k_fcconv<<<(total + 255) / 256, 256, 0, stream>>>(ws + o_hw, ws + o_hd, fc_w, fc_b, ws + o_ht);
    k_fcconv<<<(total + 255) / 256, 256, 0, stream>>>(ws + o_cw, ws + o_cd, fc_w, fc_b, ws + o_ct);
  }
  run_gclstm(ws + o_cbt, ws + o_hst, ws + o_ht, ws + o_ct);

  // ---- emb conv1d GEMMs (WMMA) ----
  k_embgemm<<<384, 256, 0, stream>>>(emb_w, emb_b, ws + o_hst, ws + o_qin);
  k_embgemm<<<384, 256, 0, stream>>>(emb_w, emb_b, ws + o_hsd, ws + o_din);
  k_embgemm<<<384, 256, 0, stream>>>(emb_w, emb_b, ws + o_hsw, ws + o_win);

  // ---- attention projections (WMMA) ----
  auto proj = [&](const float* X, int K, const float* Wm, int M, const float* bias,
                  const float* res, int resLd, float* dst, int outLd, int doRelu, int perBDst,
                  int segOff) {
    int MT = (M + 15) / 16, JT = kJ / 16;
    int blocks = (MT * JT + 7) / 8;
    if (K == 256)
      k_proj<256><<<blocks, 256, 0, stream>>>(X, Wm, M, bias, res, resLd, dst, outLd, doRelu,
                                              perBDst, segOff);
    else
      k_proj<128><<<blocks, 256, 0, stream>>>(X, Wm, M, bias, res, resLd, dst, outLd, doRelu,
                                              perBDst, segOff);
  };
  proj(ws + o_qin, 256, aq_w, 256, aq_b, nullptr, 0, ws + o_q, 256, 0, 12, 0);
  proj(ws + o_qin, 256, ak_w, 256, ak_b, nullptr, 0, ws + o_k, 256, 0, 36, 0);
  proj(ws + o_din, 256, ak_w, 256, ak_b, nullptr, 0, ws + o_k, 256, 0, 36, 12);
  proj(ws + o_win, 256, ak_w, 256, ak_b, nullptr, 0, ws + o_k, 256, 0, 36, 24);
  proj(ws + o_qin, 256, av_w, 256, av_b, nullptr, 0, ws + o_v, 256, 0, 36, 0);
  proj(ws + o_din, 256, av_w, 256, av_b, nullptr, 0, ws + o_v, 256, 0, 36, 12);
  proj(ws + o_win, 256, av_w, 256, av_b, nullptr, 0, ws + o_v, 256, 0, 36, 24);

  k_attn<<<kB * kNH, 128, 0, stream>>>(ws + o_q, ws + o_k, ws + o_v, ws + o_ctx);

  proj(ws + o_ctx, 256, ao_w, 256, ao_b, nullptr, 0, ws + o_ctx2, 256, 0, 12, 0);
  proj(ws + o_ctx2, 256, f1_w, 128, nullptr, nullptr, 0, ws + o_t1, 128, 1, 12, 0);
  // f2 + residual(ctx2) -> reuse o_ctx
  proj(ws + o_t1, 128, f2_w, 256, nullptr, ws + o_ctx2, 256, ws + o_ctx, 256, 0, 12, 0);
  k_ln<<<384, 256, 0, stream>>>(ws + o_ctx, ws + o_ctx3);
  // final: relu(ctx3 @ l1_w.T + l1_b) + Complete_OD  -> d_out
  proj(ws + o_ctx3, 256, l1_w, kNF, l1_b, COD, kNF, outp, kNF, 1, 12, 0);
}